// ItemCollaborativeFiltering_22136261443918
// MI455X (gfx1250) — compile-verified
//
#include <hip/hip_runtime.h>
#include <hip/hip_bf16.h>
#include <stdint.h>
#include <stddef.h>

// Problem constants (match reference)
#define T_DIM 64
#define U_DIM 339
#define I_DIM 5825
#define K_SEL 50
#define NTHREADS 256
#define NPT ((I_DIM + NTHREADS - 1) / NTHREADS) // 23

// CDNA5 async global->LDS path (gfx1250). Guarded so either toolchain compiles.
#ifdef __HIP_DEVICE_COMPILE__
#if __has_builtin(__builtin_amdgcn_global_load_async_to_lds_b32)
#define HAVE_ASYNC_LDS 1
#endif
#endif
#ifndef HAVE_ASYNC_LDS
#define HAVE_ASYNC_LDS 0
#endif

#if HAVE_ASYNC_LDS
typedef __attribute__((address_space(1))) int GINT; // global (AS1) int
typedef __attribute__((address_space(3))) int LINT; // LDS (AS3) int
#endif

// Monotone float -> u32 key; pack with inverted index so that u64-max gives
// (largest value, then smallest index) == jax.lax.top_k tie order.
__device__ __forceinline__ unsigned long long icf_key(float v, unsigned int idx) {
  unsigned int u = __float_as_uint(v);
  u = (u & 0x80000000u) ? ~u : (u | 0x80000000u);
  return ((unsigned long long)u << 32) | (unsigned long long)(0xFFFFFFFFu - idx);
}

__global__ __launch_bounds__(NTHREADS) void icf_topk_kernel(
    const float* __restrict__ qos,       // [T,U,I]
    const float* __restrict__ item_avg,  // [T,I]
    const float* __restrict__ sim,       // [I,I]
    const int*  __restrict__ user_id,    // [B]
    const int*  __restrict__ item_id,    // [B]
    const int*  __restrict__ time_id,    // [B]
    float*      __restrict__ out)        // [B]
{
  __shared__ float s_sim[I_DIM];
  __shared__ float s_qos[I_DIM];
  __shared__ unsigned long long s_best;
  __shared__ float s_ssum;
  __shared__ float s_wsum;

  const int b   = blockIdx.x;
  const int tid = threadIdx.x;

  const int t  = time_id[b];
  const int u  = user_id[b];
  const int it = item_id[b];

  const float* __restrict__ simRow = sim + (size_t)it * I_DIM;
  const float* __restrict__ qosRow = qos + ((size_t)t * U_DIM + (size_t)u) * (size_t)I_DIM;
  const float* __restrict__ avgRow = item_avg + (size_t)t * I_DIM;

  if (tid == 0) { s_ssum = 0.0f; s_wsum = 0.0f; }

  // Warm caches for the later scattered item_avg gathers (global_prefetch_b8).
  __builtin_prefetch(avgRow + (size_t)tid * (I_DIM / NTHREADS), 0, 1);

#if HAVE_ASYNC_LDS
  // Stage both rows into LDS with the CDNA5 async DMA path (ASYNCcnt).
  // b32 per lane: row base addresses are only 4B-aligned, so wider async
  // transfers are not alignment-safe.
#pragma unroll
  for (int j = 0; j < NPT; ++j) {
    int i = tid + j * NTHREADS;
    if (i < I_DIM) {
      __builtin_amdgcn_global_load_async_to_lds_b32(
          (GINT*)(simRow + i), (LINT*)(&s_sim[i]), 0, 0);
      __builtin_amdgcn_global_load_async_to_lds_b32(
          (GINT*)(qosRow + i), (LINT*)(&s_qos[i]), 0, 0);
    }
  }
#if __has_builtin(__builtin_amdgcn_s_wait_asynccnt)
  __builtin_amdgcn_s_wait_asynccnt(0);
#else
  asm volatile("s_wait_asynccnt 0x0" ::: "memory");
#endif
#endif
  __syncthreads();

  // Pull this thread's 23 strided elements into registers; apply rating mask.
  float sv[NPT], qv[NPT];
  unsigned int used = 0u; // bit j set => consumed or out-of-range
#pragma unroll
  for (int j = 0; j < NPT; ++j) {
    int i = tid + j * NTHREADS;
    if (i < I_DIM) {
#if HAVE_ASYNC_LDS
      float q = s_qos[i];
      float s = s_sim[i];
#else
      float q = qosRow[i];
      float s = simRow[i];
#endif
      qv[j] = q;
      sv[j] = (q > 0.0f) ? s : 0.0f; // jnp.where(user_rated, sim, 0.0)
    } else {
      qv[j] = 0.0f;
      sv[j] = 0.0f;
      used |= (1u << j);
    }
  }

  // Thread-local running argmax over unconsumed elements.
  unsigned long long bestKey = 0ull;
  int bestJ = -1, bestI = 0;
  float bestV = 0.0f, bestQ = 0.0f;

  auto recompute = [&]() {
    bestKey = 0ull;
    bestJ = -1;
#pragma unroll
    for (int j = 0; j < NPT; ++j) {
      if (!(used & (1u << j))) {
        int i = tid + j * NTHREADS;
        unsigned long long k = icf_key(sv[j], (unsigned int)i);
        if (k > bestKey) {
          bestKey = k; bestJ = j; bestV = sv[j]; bestQ = qv[j]; bestI = i;
        }
      }
    }
  };
  recompute();

  // Extract top-K: wave32 shuffle max-reduce + ds_max_u64 across the 8 waves.
  for (int k = 0; k < K_SEL; ++k) {
    if (tid == 0) s_best = 0ull;
    __syncthreads();

    unsigned long long r = bestKey;
#pragma unroll
    for (int off = 16; off > 0; off >>= 1) {
      unsigned long long o = __shfl_xor(r, off, 32);
      if (o > r) r = o;
    }
    if ((tid & 31) == 0) atomicMax(&s_best, r); // ds_max_u64
    __syncthreads();

    // Unique winner (keys embed the element index) contributes and rescans.
    // Exactly one contribution per round, barrier-separated => deterministic.
    if (bestJ >= 0 && s_best == bestKey) {
      float a = avgRow[bestI];              // item_avg[t, top_item]
      atomicAdd(&s_ssum, bestV);            // ds_add_f32
      atomicAdd(&s_wsum, bestV * (bestQ - a));
      used |= (1u << bestJ);
      recompute();
    }
    __syncthreads();
  }

  if (tid == 0) {
    float S = s_ssum;
    float W = s_wsum;
    // pred = avg_i + sum(top_sim/(S+1e-8) * (r_sf - avg_f))
    out[b] = avgRow[it] + W / (S + 1e-8f);
  }
}

extern "C" void kernel_launch(void* const* d_in, const int* in_sizes, int n_in,
                              void* d_out, int out_size, void* d_ws, size_t ws_size,
                              hipStream_t stream) {
  (void)n_in; (void)d_ws; (void)ws_size; (void)out_size;
  const float* qos      = (const float*)d_in[0]; // qos_tensor [T,U,I]
  const float* item_avg = (const float*)d_in[1]; // [T,I]
  const float* sim      = (const float*)d_in[2]; // item_sim_agg [I,I]
  const int*   user_id  = (const int*)d_in[3];
  const int*   item_id  = (const int*)d_in[4];
  const int*   time_id  = (const int*)d_in[5];
  float* out = (float*)d_out;

  const int B = in_sizes[3]; // batch size from user_id
  icf_topk_kernel<<<dim3(B), dim3(NTHREADS), 0, stream>>>(
      qos, item_avg, sim, user_id, item_id, time_id, out);
}